// TwoLNWithDropout_85839216378290
// MI455X (gfx1250) — compile-verified
//
#include <hip/hip_runtime.h>

typedef float v2f __attribute__((ext_vector_type(2)));
typedef float v8f __attribute__((ext_vector_type(8)));

#define LN_EPS 1e-5f
#define DIM 16
#define LDS_STRIDE 20   // padded row stride (dwords): rows r and r+8 land 32 banks apart

// lane-pair swap (lanes L <-> L+16): ds_swizzle group-of-32, xor=0x10, and=0x1f
__device__ __forceinline__ float swap16(float v) {
    return __int_as_float(__builtin_amdgcn_ds_swizzle(__float_as_int(v), 0x401F));
}

__global__ void __launch_bounds__(256) twoln_fc_dropout_kernel(
    const float* __restrict__ x,
    const float* __restrict__ ln1_w, const float* __restrict__ ln1_b,
    const float* __restrict__ fc_w,  const float* __restrict__ fc_b,
    const float* __restrict__ ln2_w, const float* __restrict__ ln2_b,
    const float* __restrict__ mask,
    float* __restrict__ out,
    int numTiles)
{
    __shared__ float ldsT[8 * 16 * LDS_STRIDE];   // 1.25 KB per wave, 10 KB per block

    const int lane = threadIdx.x & 31;
    const int wave = threadIdx.x >> 5;
    const int wavesPerBlock = blockDim.x >> 5;
    const int globalWave = blockIdx.x * wavesPerBlock + wave;
    const int totalWaves = gridDim.x * wavesPerBlock;

    const int half = lane >> 4;   // lane half: owns K sub-pair (A side) / row group (C side)
    const int l16  = lane & 15;   // row (A side) / output column (C side)

    float* myLds = ldsT + wave * (16 * LDS_STRIDE);

    // ---- per-lane constants --------------------------------------------------
    // K indices this lane owns in chunk c: k = 4c + 2*half, 4c + 2*half + 1
    v2f ln1wv[4], ln1bv[4], ln2wv[4], ln2bv[4], bmat[4];
#pragma unroll
    for (int c = 0; c < 4; ++c) {
        const int k = 4 * c + 2 * half;
        ln1wv[c] = *(const v2f*)(ln1_w + k);
        ln1bv[c] = *(const v2f*)(ln1_b + k);
        ln2wv[c] = *(const v2f*)(ln2_w + k);
        ln2bv[c] = *(const v2f*)(ln2_b + k);
        bmat[c]  = *(const v2f*)(fc_w + l16 * DIM + k);   // B[k][n] = fc_w[n][k]
    }
    const float fcb = fc_b[l16];                          // per-output-column bias

    for (int tile = globalWave; tile < numTiles; tile += totalWaves) {
        const int base = tile * (16 * DIM);               // tile start (elements)
        const int rowOff = base + l16 * DIM + 2 * half;   // this lane's row, K-phase

        // ---- load 8 elems of our row (WMMA A-chunk layout), 4x b64 ----------
        v2f hv[4];
#pragma unroll
        for (int c = 0; c < 4; ++c)
            hv[c] = *(const v2f*)(x + rowOff + 4 * c);

        // ---- LayerNorm 1: in-lane partials + partner-lane combine -----------
        float s = 0.f, ss = 0.f;
#pragma unroll
        for (int c = 0; c < 4; ++c) {
            s  += hv[c].x + hv[c].y;
            ss += hv[c].x * hv[c].x + hv[c].y * hv[c].y;
        }
        s  += swap16(s);
        ss += swap16(ss);
        const float mu  = s * (1.0f / 16.0f);
        const float var = ss * (1.0f / 16.0f) - mu * mu;
        const float rs  = rsqrtf(var + LN_EPS);

        v2f a[4];
#pragma unroll
        for (int c = 0; c < 4; ++c) {
            a[c].x = (hv[c].x - mu) * rs * ln1wv[c].x + ln1bv[c].x;
            a[c].y = (hv[c].y - mu) * rs * ln1wv[c].y + ln1bv[c].y;
        }

        // ---- 16x16x16 fp32 GEMM tile: 4 chained V_WMMA_F32_16X16X4_F32 ------
        v8f acc = {};
#pragma unroll
        for (int c = 0; c < 4; ++c)
            acc = __builtin_amdgcn_wmma_f32_16x16x4_f32(
                false, a[c], false, bmat[c], (short)0, acc, false, false);

        // ---- transpose D tile through wave-private LDS ----------------------
        // C-layout: lane holds col=l16 of rows (r + 8*half). Add fc_b here.
#pragma unroll
        for (int r = 0; r < 8; ++r)
            myLds[(r + 8 * half) * LDS_STRIDE + l16] = acc[r] + fcb;

        // read back in row-per-lane-pair layout (same as input layout)
        v2f hp[4];
#pragma unroll
        for (int c = 0; c < 4; ++c)
            hp[c] = *(const v2f*)(myLds + l16 * LDS_STRIDE + 4 * c + 2 * half);

        // ---- dropout mask (coalesced b64 loads) + ReLU ----------------------
        float s2 = 0.f, q2 = 0.f;
#pragma unroll
        for (int c = 0; c < 4; ++c) {
            const v2f mv = *(const v2f*)(mask + rowOff + 4 * c);
            hp[c].x = fmaxf(hp[c].x * mv.x, 0.0f);
            hp[c].y = fmaxf(hp[c].y * mv.y, 0.0f);
            s2 += hp[c].x + hp[c].y;
            q2 += hp[c].x * hp[c].x + hp[c].y * hp[c].y;
        }

        // ---- LayerNorm 2: same cheap reduction as LN1 -----------------------
        s2 += swap16(s2);
        q2 += swap16(q2);
        const float mu2  = s2 * (1.0f / 16.0f);
        const float var2 = q2 * (1.0f / 16.0f) - mu2 * mu2;
        const float rs2  = rsqrtf(var2 + LN_EPS);

        // ---- scale/shift + coalesced b64 stores -----------------------------
#pragma unroll
        for (int c = 0; c < 4; ++c) {
            v2f o;
            o.x = (hp[c].x - mu2) * rs2 * ln2wv[c].x + ln2bv[c].x;
            o.y = (hp[c].y - mu2) * rs2 * ln2wv[c].y + ln2bv[c].y;
            *(v2f*)(out + rowOff + 4 * c) = o;
        }
    }
}

extern "C" void kernel_launch(void* const* d_in, const int* in_sizes, int n_in,
                              void* d_out, int out_size, void* d_ws, size_t ws_size,
                              hipStream_t stream) {
    const float* x     = (const float*)d_in[0];
    const float* ln1_w = (const float*)d_in[1];
    const float* ln1_b = (const float*)d_in[2];
    const float* fc_w  = (const float*)d_in[3];
    const float* fc_b  = (const float*)d_in[4];
    const float* ln2_w = (const float*)d_in[5];
    const float* ln2_b = (const float*)d_in[6];
    const float* mask  = (const float*)d_in[7];
    float* out = (float*)d_out;

    const int rows = in_sizes[0] / DIM;       // 4194304
    const int numTiles = rows / 16;           // 262144

    const int threads = 256;                  // 8 wave32 per block
    const int wavesPerBlock = threads / 32;
    int blocks = (numTiles + wavesPerBlock - 1) / wavesPerBlock;
    if (blocks > 4096) blocks = 4096;         // persistent waves, grid-stride over tiles

    twoln_fc_dropout_kernel<<<blocks, threads, 0, stream>>>(
        x, ln1_w, ln1_b, fc_w, fc_b, ln2_w, ln2_b, mask, out, numTiles);
}